// RoIPool_14748917694865
// MI455X (gfx1250) — compile-verified
//
#include <hip/hip_runtime.h>
#include <math.h>

// RoIPool forward, MI455X (gfx1250, wave32).
// features: (N, C=256, H=32, W=32) fp32;  rois: (R, 5) fp32
// out: (R, C, 7, 7) fp32
//
// Pipeline (all on `stream`, graph-capture safe):
//   K0: zero per-batch roi counters in workspace.
//   K1: per-ROI quantized bin bounds -> bnd[r*32] (32 ints:
//       [0]=batch, [1..7]=hs, [8..14]=he, [15..21]=ws, [22..28]=we,
//       [29]=global roi id), then atomic scatter roi id into its
//       batch's compacted list (order nondeterministic; output is
//       order-invariant since each roi owns a disjoint out slice).
//   K2: block = (plane b*C+c, 128-roi list split). Stage 4KB feature
//       plane + gathered 16KB bound rows into LDS; 49 lanes per ROI
//       (ph,pw) max-reduce their bin entirely from LDS. Output stores
//       are non-temporal (write-once stream to HBM).

#define OUTP 7
#define POOL2 49            // 7*7
#define RSCALE 0.0625f
#define CFIX 256
#define HFIX 32
#define WFIX 32
#define PLANE_ELEMS (HFIX * WFIX)      // 1024 floats = 4KB
#define ROIS_PER_SPLIT 128
#define ROIS_PER_ITER 5                // 5*49 = 245 active lanes of 256
#define NEG_FLT (-3.402823466e38f)

__global__ void roipool_zero_kernel(unsigned int* __restrict__ counters, int n)
{
    int i = blockIdx.x * blockDim.x + threadIdx.x;
    if (i < n) counters[i] = 0u;
}

__global__ __launch_bounds__(256) void roipool_bounds_kernel(
    const float* __restrict__ rois, int* __restrict__ bnd,
    unsigned int* __restrict__ counters, int* __restrict__ lists,
    int R, int P)
{
    int r = blockIdx.x * blockDim.x + threadIdx.x;
    if (r >= R) return;

    const float* roi = rois + (size_t)r * 5;
    int   b    = (int)roi[0];
    float rs_w = floorf(roi[1] * RSCALE + 0.5f);
    float rs_h = floorf(roi[2] * RSCALE + 0.5f);
    float re_w = floorf(roi[3] * RSCALE + 0.5f);
    float re_h = floorf(roi[4] * RSCALE + 0.5f);
    float roi_w = fmaxf(re_w - rs_w + 1.0f, 1.0f);
    float roi_h = fmaxf(re_h - rs_h + 1.0f, 1.0f);
    float bin_w = roi_w / (float)OUTP;
    float bin_h = roi_h / (float)OUTP;

    int* B = bnd + (size_t)r * 32;
    B[0] = b;
    #pragma unroll
    for (int p = 0; p < OUTP; ++p) {
        float pf = (float)p;
        float hs = fminf(fmaxf(floorf(pf * bin_h) + rs_h, 0.0f), (float)HFIX);
        float he = fminf(fmaxf(ceilf((pf + 1.0f) * bin_h) + rs_h, 0.0f), (float)HFIX);
        float ws = fminf(fmaxf(floorf(pf * bin_w) + rs_w, 0.0f), (float)WFIX);
        float we = fminf(fmaxf(ceilf((pf + 1.0f) * bin_w) + rs_w, 0.0f), (float)WFIX);
        B[1  + p] = (int)hs;
        B[8  + p] = (int)he;
        B[15 + p] = (int)ws;
        B[22 + p] = (int)we;
    }
    B[29] = r;
    B[30] = 0;
    B[31] = 0;

    unsigned int slot = atomicAdd(&counters[b], 1u);
    lists[(size_t)b * P + slot] = r;
}

__global__ __launch_bounds__(256) void roipool_pool_kernel(
    const float* __restrict__ features, const int* __restrict__ bnd,
    const unsigned int* __restrict__ counters, const int* __restrict__ lists,
    float* __restrict__ out, int P)
{
    __shared__ float plane[PLANE_ELEMS];             // 4 KB
    __shared__ int   sbnd[ROIS_PER_SPLIT * 32];      // 16 KB

    const int plane_id = blockIdx.x;                 // b*C + c
    const int split    = blockIdx.y;
    const int b        = plane_id / CFIX;
    const int c        = plane_id % CFIX;
    const int t        = threadIdx.x;

    const int cnt  = (int)counters[b];               // rois in this batch
    const int base = split * ROIS_PER_SPLIT;
    if (base >= cnt) return;                         // block-uniform exit
    const int nvalid = min(ROIS_PER_SPLIT, cnt - base);

    // --- Stage the 4KB feature plane into LDS (float4 per thread) ---
    const float4* src4 = (const float4*)(features + (size_t)plane_id * PLANE_ELEMS);
    ((float4*)plane)[t] = src4[t];

    // --- Gather this split's listed roi bound-rows (128B each) into LDS ---
    const int* mylist = lists + (size_t)b * P + base;
    int4* s4 = (int4*)sbnd;
    #pragma unroll
    for (int i = 0; i < 4; ++i) {
        int idx  = t + i * 256;                      // int4 index, 8 per roi
        int rloc = idx >> 3;
        int part = idx & 7;
        if (rloc < nvalid) {
            int g_roi = mylist[rloc];
            s4[idx] = ((const int4*)(bnd + (size_t)g_roi * 32))[part];
        }
    }

    __syncthreads();   // s_barrier_signal/s_barrier_wait (wave32 WG barrier)

    const bool active = (t < ROIS_PER_ITER * POOL2);  // 245 lanes
    const int  lroi   = t / POOL2;
    const int  pos    = t - lroi * POOL2;
    const int  ph     = pos / OUTP;
    const int  pw     = pos - ph * OUTP;

    for (int bl = 0; bl < nvalid; bl += ROIS_PER_ITER) {
        int rl = bl + lroi;                           // local roi in split
        if (active && rl < nvalid) {
            const int* B = sbnd + rl * 32;            // guaranteed batch match
            int hs = B[1  + ph];
            int he = B[8  + ph];
            int ws = B[15 + pw];
            int we = B[22 + pw];
            int r  = B[29];                           // global roi id
            bool empty = (he <= hs) || (we <= ws);
            float m = NEG_FLT;
            for (int h = hs; h < he; ++h) {
                const float* row = plane + h * WFIX;
                for (int w = ws; w < we; ++w)
                    m = fmaxf(m, row[w]);             // ds_load_b32 + v_max
            }
            // write-once output: stream past caches (TH_NT store)
            __builtin_nontemporal_store(
                empty ? 0.0f : m,
                &out[((size_t)r * CFIX + c) * POOL2 + pos]);
        }
    }
}

extern "C" void kernel_launch(void* const* d_in, const int* in_sizes, int n_in,
                              void* d_out, int out_size, void* d_ws, size_t ws_size,
                              hipStream_t stream) {
    const float* features = (const float*)d_in[0];
    const float* rois     = (const float*)d_in[1];
    float*       out      = (float*)d_out;

    const int R = in_sizes[1] / 5;
    const int N = in_sizes[0] / (CFIX * HFIX * WFIX);

    const int nsplit = (R + ROIS_PER_SPLIT - 1) / ROIS_PER_SPLIT;
    const int P      = nsplit * ROIS_PER_SPLIT;     // list stride per batch

    // workspace layout (16B-aligned base from hipMalloc):
    //   [0)            bnd      : P*32 ints
    //   [P*32)         counters : N uints
    //   [+pad to 16B)  lists    : N*P ints
    int*          bnd      = (int*)d_ws;
    unsigned int* counters = (unsigned int*)(bnd + (size_t)P * 32);
    int*          lists    = (int*)(counters + ((N + 3) & ~3));  // keep 16B align

    roipool_zero_kernel<<<1, 64, 0, stream>>>(counters, N);

    roipool_bounds_kernel<<<(R + 255) / 256, 256, 0, stream>>>(
        rois, bnd, counters, lists, R, P);

    dim3 grid(N * CFIX, nsplit);                    // (512, 4) @ reference shapes
    roipool_pool_kernel<<<grid, 256, 0, stream>>>(
        features, bnd, counters, lists, out, P);
}